// SelfAttention_19387482374279
// MI455X (gfx1250) — compile-verified
//
#include <hip/hip_runtime.h>
#include <math.h>

// ---------------------------------------------------------------------------
// Self-attention (B=4, N=8192, d=3) for gfx1250 (MI455X, wave32, WMMA).
// Phase 1: QKV projection -> padded float4 workspace (q scaled by 1/sqrt(3)).
// Phase 2: flash attention, one 16-query tile per wave, K/V streamed through
//          LDS in 256-key chunks, S^T = K @ Q^T via V_WMMA_F32_16X16X4_F32,
//          online softmax with a single cross-half shfl_xor(16) per tile,
//          P@V (3 features) done with VALU FMAs against LDS V.
// ---------------------------------------------------------------------------

typedef float v2f __attribute__((ext_vector_type(2)));
typedef float v8f __attribute__((ext_vector_type(8)));

#define BB 4
#define NN 8192
#define CHUNK 256   // keys staged in LDS per block iteration
#define QTILE 16    // queries per wave
#define WAVES 8     // waves per block (256 threads)

// ---------------- Phase 1: QKV projection -----------------------------------
// Reference: qkv = x @ W^T + b, view (B,N,3,3); q = [...,0] -> channels 0,3,6
//                                               k = [...,1] -> channels 1,4,7
//                                               v = [...,2] -> channels 2,5,8
__global__ __launch_bounds__(256) void qkv_proj_kernel(
    const float* __restrict__ x, const float* __restrict__ W,
    const float* __restrict__ bias,
    float4* __restrict__ qws, float4* __restrict__ kws, float4* __restrict__ vws)
{
    const int idx = blockIdx.x * blockDim.x + threadIdx.x;   // 0 .. B*N-1
    const float x0 = x[idx * 3 + 0];
    const float x1 = x[idx * 3 + 1];
    const float x2 = x[idx * 3 + 2];

    float ch[9];
#pragma unroll
    for (int o = 0; o < 9; ++o) {
        ch[o] = fmaf(x0, W[o * 3 + 0],
                fmaf(x1, W[o * 3 + 1],
                fmaf(x2, W[o * 3 + 2], bias[o])));
    }
    const float scale = 0.57735026918962576451f;  // 1/sqrt(3), folded into q
    qws[idx] = make_float4(ch[0] * scale, ch[3] * scale, ch[6] * scale, 0.0f);
    kws[idx] = make_float4(ch[1], ch[4], ch[7], 0.0f);
    vws[idx] = make_float4(ch[2], ch[5], ch[8], 0.0f);
}

// ---------------- Phase 2: flash attention -----------------------------------
__global__ __launch_bounds__(256) void attn_kernel(
    const float* __restrict__ x,
    const float4* __restrict__ qws, const float4* __restrict__ kws,
    const float4* __restrict__ vws, float* __restrict__ out)
{
    __shared__ float4 k_s[CHUNK];
    __shared__ float4 v_s[CHUNK];

    const int b    = blockIdx.y;
    const int lane = threadIdx.x & 31;
    const int wave = threadIdx.x >> 5;
    const int half = lane >> 4;                  // 0: components {0,1} / keys 0-7
                                                 // 1: components {2,3} / keys 8-15
    const int comp  = half * 2;
    const int khalf = half * 8;
    const int qrow  = blockIdx.x * (WAVES * QTILE) + wave * QTILE + (lane & 15);

    const float4* kbase = kws + (size_t)b * NN;
    const float4* vbase = vws + (size_t)b * NN;

    // WMMA B operand (Q^T, 4x16): lane l<16 -> q[l][{0,1}], lane l>=16 -> q[l-16][{2,3}]
    v2f bq;
    {
        const float* qp = (const float*)(qws + (size_t)b * NN + qrow);
        bq.x = qp[comp];
        bq.y = qp[comp + 1];
    }

    float m = -INFINITY;        // running max for this lane's query
    float l = 0.0f;             // running softmax denominator (this half's keys)
    float acc0 = 0.0f, acc1 = 0.0f, acc2 = 0.0f;

    for (int c0 = 0; c0 < NN; c0 += CHUNK) {
        __syncthreads();
        {
            const int key = c0 + (int)threadIdx.x;   // 256 threads, one key each
            k_s[threadIdx.x] = kbase[key];
            v_s[threadIdx.x] = vbase[key];
            if (c0 + CHUNK < NN) {                   // gfx1250 global_prefetch_b8
                __builtin_prefetch(&kbase[key + CHUNK], 0, 0);
                __builtin_prefetch(&vbase[key + CHUNK], 0, 0);
            }
        }
        __syncthreads();

        for (int t = 0; t < CHUNK; t += QTILE) {
            // WMMA A operand (K tile, 16x4): lane l<16 -> k[t+l][{0,1}],
            //                                lane l>=16 -> k[t+l-16][{2,3}]
            v2f ak;
            {
                const float* kp = (const float*)&k_s[t + (lane & 15)];
                ak.x = kp[comp];
                ak.y = kp[comp + 1];
            }
            v8f c = {};
            // S^T = K_tile(16x4) x Q^T(4x16); D VGPR r: lane l<16 -> (key r, query l),
            // lane l>=16 -> (key 8+r, query l-16).  Scale already folded into q.
            v8f s = __builtin_amdgcn_wmma_f32_16x16x4_f32(
                false, ak, false, bq, (short)0, c, false, false);

            // --- online softmax over this tile's 16 keys (8 local + 8 in pair lane)
            float tmax = s[0];
#pragma unroll
            for (int j = 1; j < 8; ++j) tmax = fmaxf(tmax, s[j]);
            tmax = fmaxf(tmax, __shfl_xor(tmax, 16, 32));
            const float newm = fmaxf(m, tmax);
            const float corr = __expf(m - newm);
            l *= corr; acc0 *= corr; acc1 *= corr; acc2 *= corr;

#pragma unroll
            for (int j = 0; j < 8; ++j) {
                const float p = __expf(s[j] - newm);
                l += p;
                const float4 v4 = v_s[t + khalf + j];
                acc0 = fmaf(p, v4.x, acc0);
                acc1 = fmaf(p, v4.y, acc1);
                acc2 = fmaf(p, v4.z, acc2);
            }
            m = newm;
        }
    }

    // Merge the two half-lanes (same query, disjoint key halves, same m).
    l    += __shfl_xor(l,    16, 32);
    acc0 += __shfl_xor(acc0, 16, 32);
    acc1 += __shfl_xor(acc1, 16, 32);
    acc2 += __shfl_xor(acc2, 16, 32);

    if (lane < 16) {
        const float invl = 1.0f / l;
        const size_t o = ((size_t)b * NN + qrow) * 3;
        out[o + 0] = fmaf(acc0, invl, x[o + 0]);
        out[o + 1] = fmaf(acc1, invl, x[o + 1]);
        out[o + 2] = fmaf(acc2, invl, x[o + 2]);
    }
}

// ---------------------------------------------------------------------------
extern "C" void kernel_launch(void* const* d_in, const int* in_sizes, int n_in,
                              void* d_out, int out_size, void* d_ws, size_t ws_size,
                              hipStream_t stream)
{
    const float* x    = (const float*)d_in[0];   // (B, N, 3)
    const float* Wqkv = (const float*)d_in[1];   // (9, 3)
    const float* bqkv = (const float*)d_in[2];   // (9,)
    float* out = (float*)d_out;                  // (B, N, 3)

    // Workspace: q/k/v padded to float4, B*N rows each (512 KB per tensor).
    float4* qws = (float4*)d_ws;
    float4* kws = qws + (size_t)BB * NN;
    float4* vws = kws + (size_t)BB * NN;

    qkv_proj_kernel<<<dim3((BB * NN) / 256), dim3(256), 0, stream>>>(
        x, Wqkv, bqkv, qws, kws, vws);

    attn_kernel<<<dim3(NN / (WAVES * QTILE), BB), dim3(256), 0, stream>>>(
        x, qws, kws, vws, out);
}